// Attention_42279658062639
// MI455X (gfx1250) — compile-verified
//
#include <hip/hip_runtime.h>

// Topic-modulated attention forward for MI455X (gfx1250, wave32).
//
// Roofline: output p_attn (268 MB) dominates HBM traffic (~285 MB total at
// 23.3 TB/s ~= 12-14 us); K/V working set (33 MB) is L2-resident (192 MB L2),
// so per-q-tile K/V re-reads are cheap. Memory-bound => keep fp32, use
// V_WMMA_F32_16X16X4_F32 for QK / topicQK / PV, stage K/V tiles into LDS with
// async copy-to-LDS (ASYNCcnt) when available, and use non-temporal stores
// for the p_attn stream so it does not evict L2-resident K/V.

#define Bsz  4
#define Hn   16
#define Sl   1024
#define Dh   64
#define NEGV (-1000000000.0f)
#define SCALE 0.125f  /* 1/sqrt(64) for both score scalings */

// K/V staging tile: 16 rows x 64 cols, padded to 68 (row stride 272 B = 17*16B:
// keeps b128 stores 16B-aligned and makes frag reads bank-conflict-free).
#define KVP 68

typedef float v2f __attribute__((ext_vector_type(2)));
typedef float v8f __attribute__((ext_vector_type(8)));
typedef int   v4i __attribute__((ext_vector_type(4)));

#define AS1 __attribute__((address_space(1)))
#define AS3 __attribute__((address_space(3)))

#if defined(__has_builtin)
#if __has_builtin(__builtin_amdgcn_global_load_async_to_lds_b128) && \
    __has_builtin(__builtin_amdgcn_s_wait_asynccnt)
#define HAVE_ASYNC_LDS 1
#endif
#endif

// Stage a 16x64 fp32 tile (row stride 64 floats in global) into LDS with
// padded row stride KVP. One wave, 8 x b128 per lane-set, fully coalesced
// (512 B per instruction).
__device__ __forceinline__ void stage16x64(const float* __restrict__ g,
                                           float* lds, int lane)
{
    #pragma unroll
    for (int it = 0; it < 8; ++it) {
        const int flat = it * 32 + lane;   // 256 float4 slots
        const int row  = flat >> 4;        // 16 float4 per row
        const int c4   = flat & 15;
        const float* gp = g + row * 64 + c4 * 4;
        float*       lp = lds + row * KVP + c4 * 4;
#if defined(HAVE_ASYNC_LDS)
        AS1 v4i* g1 = (AS1 v4i*)(v4i*)gp;   // reinterpret, then generic->global
        AS3 v4i* l1 = (AS3 v4i*)(v4i*)lp;   // reinterpret, then generic->LDS
        __builtin_amdgcn_global_load_async_to_lds_b128(g1, l1, 0, 0);
#else
        const float4 t = *(const float4*)gp;
        *(float4*)lp = t;
#endif
    }
#if defined(HAVE_ASYNC_LDS)
    __builtin_amdgcn_s_wait_asynccnt(0);
    __asm__ volatile("" ::: "memory");
#endif
}

__global__ __launch_bounds__(32)
void attn_topic_kernel(const float* __restrict__ Q, const float* __restrict__ K,
                       const float* __restrict__ V, const int* __restrict__ mask,
                       const float* __restrict__ TQ, const float* __restrict__ TK,
                       float* __restrict__ outO, float* __restrict__ outP)
{
    // 16 q-rows x 1024 k-cols of fp32 scores: 64 KB LDS.
    __shared__ float sS[16 * Sl];
    // K / topicK / V tile staging buffer (4.25 KB).
    __shared__ float sKV[16 * KVP];

    const int lane = threadIdx.x;       // 0..31 (wave32)
    const int half = lane >> 4;         // 0 or 1
    const int l16  = lane & 15;
    const int koff = half * 2;          // A/B frag K sub-offset per ISA layout

    const int blk = blockIdx.x;         // B*H*(S/16) = 4096 blocks
    const int b   = blk >> 10;
    const int h   = (blk >> 6) & (Hn - 1);
    const int q0  = (blk & 63) * 16;

    const float* Qh  = Q  + (size_t)(b * Hn + h) * Sl * Dh;
    const float* Kh  = K  + (size_t)(b * Hn + h) * Sl * Dh;
    const float* Vh  = V  + (size_t)(b * Hn + h) * Sl * Dh;
    const float* TQb = TQ + (size_t)b * Sl * Dh;
    const float* TKb = TK + (size_t)b * Sl * Dh;
    const int*   Mb  = mask + b * Sl;

    const v8f z8 = {0.f,0.f,0.f,0.f,0.f,0.f,0.f,0.f};

    // q-row mask bits (16 rows -> 16 bits, same in every lane)
    unsigned mqbits = 0u;
    #pragma unroll
    for (int r = 0; r < 16; ++r)
        mqbits |= (Mb[q0 + r] != 0 ? 1u : 0u) << r;

    // -------- Phase 1: topic scores (16 x 1024) -> LDS (masked -> 1.0) -----
    {
        v2f ta[16];                      // topicQ A-frags: 16 K-chunks of 4
        #pragma unroll
        for (int c = 0; c < 16; ++c) {
            const float* p = TQb + (size_t)(q0 + l16) * Dh + c * 4 + koff;
            ta[c].x = p[0]; ta[c].y = p[1];
        }
        for (int kt = 0; kt < 64; ++kt) {
            const int k0   = kt * 16;
            const int kcol = k0 + l16;          // this lane's N column
            stage16x64(TKb + (size_t)k0 * Dh, sKV, lane);
            if (kt + 1 < 64)
                __builtin_prefetch(TKb + (size_t)(k0 + 16) * Dh + lane * 32, 0, 1);
            const int mk = (Mb[kcol] != 0);
            v8f acc = z8;
            #pragma unroll
            for (int c = 0; c < 16; ++c) {
                const float* lp = sKV + l16 * KVP + c * 4 + koff;
                v2f bb; bb.x = lp[0]; bb.y = lp[1];
                acc = __builtin_amdgcn_wmma_f32_16x16x4_f32(
                        false, ta[c], false, bb, (short)0, acc, false, false);
            }
            #pragma unroll
            for (int i = 0; i < 8; ++i) {
                const int row = i + half * 8;   // C/D layout: VGPR i -> M=i / i+8
                const bool pm = ((mqbits >> row) & 1u) && mk;
                sS[row * Sl + kcol] = pm ? acc[i] * SCALE : 1.0f;
            }
        }
    }

    // -------- Phase 2: topic-row softmax stats (lane l16 <-> row l16) ------
    float mt, st;
    {
        const float* rowp = &sS[l16 * Sl + half * 512];
        float m = -3.402823466e38f;
        for (int j = 0; j < 512; ++j) m = fmaxf(m, rowp[j]);
        m = fmaxf(m, __shfl_xor(m, 16, 32));
        float s = 0.f;
        for (int j = 0; j < 512; ++j) s += __expf(rowp[j] - m);
        s += __shfl_xor(s, 16, 32);
        mt = m; st = s;
    }

    // -------- Phase 3: QK scores, mask, multiply by topic probs -> LDS -----
    {
        v2f qa[16];
        #pragma unroll
        for (int c = 0; c < 16; ++c) {
            const float* p = Qh + (size_t)(q0 + l16) * Dh + c * 4 + koff;
            qa[c].x = p[0]; qa[c].y = p[1];
        }
        float mt8[8], st8[8];
        const unsigned mq8 = (mqbits >> (half * 8)) & 0xFFu;
        #pragma unroll
        for (int i = 0; i < 8; ++i) {
            mt8[i] = __shfl(mt, i + half * 8, 32);
            st8[i] = __shfl(st, i + half * 8, 32);
        }
        for (int kt = 0; kt < 64; ++kt) {
            const int k0   = kt * 16;
            const int kcol = k0 + l16;
            stage16x64(Kh + (size_t)k0 * Dh, sKV, lane);
            if (kt + 1 < 64)
                __builtin_prefetch(Kh + (size_t)(k0 + 16) * Dh + lane * 32, 0, 1);
            const int mk = (Mb[kcol] != 0);
            v8f acc = z8;
            #pragma unroll
            for (int c = 0; c < 16; ++c) {
                const float* lp = sKV + l16 * KVP + c * 4 + koff;
                v2f bb; bb.x = lp[0]; bb.y = lp[1];
                acc = __builtin_amdgcn_wmma_f32_16x16x4_f32(
                        false, qa[c], false, bb, (short)0, acc, false, false);
            }
            #pragma unroll
            for (int i = 0; i < 8; ++i) {
                const int row = i + half * 8;
                const bool pm = ((mq8 >> i) & 1u) && mk;
                const float sc = pm ? acc[i] * SCALE : NEGV;
                const float ts = sS[row * Sl + kcol];
                const float tp = __expf(ts - mt8[i]) / st8[i];
                sS[row * Sl + kcol] = sc * tp;
            }
        }
    }

    // -------- Phase 4: attention-row softmax stats -------------------------
    float m2, s2;
    {
        const float* rowp = &sS[l16 * Sl + half * 512];
        float m = -3.402823466e38f;
        for (int j = 0; j < 512; ++j) m = fmaxf(m, rowp[j]);
        m = fmaxf(m, __shfl_xor(m, 16, 32));
        float s = 0.f;
        for (int j = 0; j < 512; ++j) s += __expf(rowp[j] - m);
        s += __shfl_xor(s, 16, 32);
        m2 = m; s2 = s;
    }

    // -------- Phase 5a: normalize; write p_attn (NT) and P back to LDS -----
    {
        float* Pout = outP + (size_t)((b * Hn + h) * Sl + q0) * Sl;
        for (int r = 0; r < 16; ++r) {
            const float m   = __shfl(m2, r, 32);
            const float inv = 1.0f / __shfl(s2, r, 32);
            for (int j = lane; j < Sl; j += 32) {
                const float p = __expf(sS[r * Sl + j] - m) * inv;
                sS[r * Sl + j] = p;
                __builtin_nontemporal_store(p, &Pout[(size_t)r * Sl + j]);
            }
        }
    }

    // -------- Phase 5b: out = P @ V via WMMA (A-frags read from LDS) -------
    v8f o0 = z8, o1 = z8, o2 = z8, o3 = z8;
    for (int kt = 0; kt < 64; ++kt) {
        const int k0 = kt * 16;
        stage16x64(Vh + (size_t)k0 * Dh, sKV, lane);
        if (kt + 1 < 64)
            __builtin_prefetch(Vh + (size_t)(k0 + 16) * Dh + lane * 32, 0, 1);
        #pragma unroll
        for (int c = 0; c < 4; ++c) {
            const int kk = k0 + c * 4 + koff;   // global k of this lane's frag
            v2f pa;                              // A = P[16 x 4 chunk]
            pa.x = sS[l16 * Sl + kk];
            pa.y = sS[l16 * Sl + kk + 1];
            const int r0 = (c * 4 + koff) * KVP;      // V rows inside sKV
            const int r1 = r0 + KVP;
            v2f bb;
            bb.x = sKV[r0 + 0 * 16 + l16]; bb.y = sKV[r1 + 0 * 16 + l16];
            o0 = __builtin_amdgcn_wmma_f32_16x16x4_f32(false, pa, false, bb, (short)0, o0, false, false);
            bb.x = sKV[r0 + 1 * 16 + l16]; bb.y = sKV[r1 + 1 * 16 + l16];
            o1 = __builtin_amdgcn_wmma_f32_16x16x4_f32(false, pa, false, bb, (short)0, o1, false, false);
            bb.x = sKV[r0 + 2 * 16 + l16]; bb.y = sKV[r1 + 2 * 16 + l16];
            o2 = __builtin_amdgcn_wmma_f32_16x16x4_f32(false, pa, false, bb, (short)0, o2, false, false);
            bb.x = sKV[r0 + 3 * 16 + l16]; bb.y = sKV[r1 + 3 * 16 + l16];
            o3 = __builtin_amdgcn_wmma_f32_16x16x4_f32(false, pa, false, bb, (short)0, o3, false, false);
        }
    }

    // -------- Final: store out tile (16 x 64) ------------------------------
    float* Oout = outO + (size_t)((b * Hn + h) * Sl + q0) * Dh;
    #pragma unroll
    for (int i = 0; i < 8; ++i) {
        const int row = i + half * 8;
        float* orow = Oout + (size_t)row * Dh;
        __builtin_nontemporal_store(o0[i], &orow[0 * 16 + l16]);
        __builtin_nontemporal_store(o1[i], &orow[1 * 16 + l16]);
        __builtin_nontemporal_store(o2[i], &orow[2 * 16 + l16]);
        __builtin_nontemporal_store(o3[i], &orow[3 * 16 + l16]);
    }
}

extern "C" void kernel_launch(void* const* d_in, const int* in_sizes, int n_in,
                              void* d_out, int out_size, void* d_ws, size_t ws_size,
                              hipStream_t stream) {
    const float* Q  = (const float*)d_in[0];
    const float* K  = (const float*)d_in[1];
    const float* V  = (const float*)d_in[2];
    /* d_in[3] = head_nums scalar (compile-time Hn) */
    const int*   M  = (const int*)d_in[4];
    const float* TQ = (const float*)d_in[5];
    const float* TK = (const float*)d_in[6];

    float* outO = (float*)d_out;                             // [B,H,S,D]
    float* outP = outO + (size_t)Bsz * Hn * Sl * Dh;         // [B,H,S,S]

    dim3 grid(Bsz * Hn * (Sl / 16));
    attn_topic_kernel<<<grid, 32, 0, stream>>>(Q, K, V, M, TQ, TK, outO, outP);
}